// Model_42803644072030
// MI455X (gfx1250) — compile-verified
//
#include <hip/hip_runtime.h>
#include <math.h>

#define SEQ_LEN 512
#define PRED_LEN 96
#define ENC_IN 32
#define D_MODEL 128
#define D_INNER 256

typedef __attribute__((ext_vector_type(2))) float v2f;
typedef __attribute__((ext_vector_type(8))) float v8f;

// ---------------------------------------------------------------------------
// Generic fp32 WMMA GEMM:  C[m,n] (+)= sum_k A[m,k] * W[n,k]  (+ bias) (+gelu)
// One wave computes a 16 x (16*NT) strip via V_WMMA_F32_16X16X4_F32, reusing
// the A fragment across NT column tiles.  Per K-chunk of 16 we first issue all
// (4 + 4*NT) independent b64 fragment loads, then the 4*NT WMMAs, so the
// scheduler can clause the loads and drain loadcnt incrementally.
// Requires: M % 16 == 0, N % (16*NT) == 0, K % 16 == 0 (true at every call).
// ---------------------------------------------------------------------------
template<int NT, int ADD_BIAS, int ACCUM, int GELU>
__global__ void __launch_bounds__(32)
gemm_wmma(const float* __restrict__ A, const float* __restrict__ W,
          const float* __restrict__ bias, float* __restrict__ C,
          int M, int N, int K) {
  const int tm   = blockIdx.x * 16;
  const int tn   = blockIdx.y * (16 * NT);
  const int lane = threadIdx.x & 31;
  const int half = lane >> 4;
  const int r    = lane & 15;
  const float* __restrict__ arow = A + (size_t)(tm + r) * K + 2 * half;
  const float* wr[NT];
#pragma unroll
  for (int t = 0; t < NT; ++t)
    wr[t] = W + (size_t)(tn + t * 16 + r) * K + 2 * half;

  v8f acc[NT];
#pragma unroll
  for (int t = 0; t < NT; ++t) acc[t] = {};

  for (int k = 0; k < K; k += 16) {
    v2f af[4];
    v2f bf[4][NT];
    // ---- issue all fragment loads for this K-chunk (independent) ----
#pragma unroll
    for (int kk = 0; kk < 4; ++kk) {
      af[kk].x = arow[k + 4 * kk];
      af[kk].y = arow[k + 4 * kk + 1];
    }
#pragma unroll
    for (int kk = 0; kk < 4; ++kk) {
#pragma unroll
      for (int t = 0; t < NT; ++t) {
        bf[kk][t].x = wr[t][k + 4 * kk];
        bf[kk][t].y = wr[t][k + 4 * kk + 1];
      }
    }
    // ---- consume: 4*NT WMMAs ----
#pragma unroll
    for (int kk = 0; kk < 4; ++kk) {
#pragma unroll
      for (int t = 0; t < NT; ++t) {
        acc[t] = __builtin_amdgcn_wmma_f32_16x16x4_f32(false, af[kk], false,
                                                       bf[kk][t], (short)0,
                                                       acc[t], false, false);
      }
    }
  }

#pragma unroll
  for (int t = 0; t < NT; ++t) {
    const int n = tn + t * 16 + r;
    const float bv = ADD_BIAS ? bias[n] : 0.0f;
#pragma unroll
    for (int v = 0; v < 8; ++v) {
      int   m   = tm + v + 8 * half;
      float val = acc[t][v] + bv;
      if (GELU) val = 0.5f * val * (1.0f + erff(val * 0.70710678118654752f));
      float* cp = C + (size_t)m * N + n;
      if (ACCUM) *cp += val; else *cp = val;
    }
  }
}

// ------------------------- RevIN statistics -------------------------------
__global__ void revin_stats(const float* __restrict__ x, float* __restrict__ mean,
                            float* __restrict__ stdv) {
  __shared__ float s1[128], s2[128];
  int bc = blockIdx.x, b = bc >> 5, c = bc & 31, t = threadIdx.x;
  float s = 0.f, q = 0.f;
  for (int l = t; l < SEQ_LEN; l += 128) {
    float v = x[((size_t)b * SEQ_LEN + l) * ENC_IN + c];
    s += v; q += v * v;
  }
  s1[t] = s; s2[t] = q; __syncthreads();
  for (int off = 64; off > 0; off >>= 1) {
    if (t < off) { s1[t] += s1[t + off]; s2[t] += s2[t + off]; }
    __syncthreads();
  }
  if (t == 0) {
    float m = s1[0] * (1.0f / SEQ_LEN);
    float var = s2[0] * (1.0f / SEQ_LEN) - m * m;
    mean[bc] = m;
    stdv[bc] = sqrtf(var + 1e-5f);
  }
}

__global__ void revin_apply(const float* __restrict__ x, const float* __restrict__ mean,
                            const float* __restrict__ stdv, const float* __restrict__ rw,
                            const float* __restrict__ rb, float* __restrict__ xcn) {
  int idx = blockIdx.x * blockDim.x + threadIdx.x;
  if (idx >= 128 * SEQ_LEN) return;
  int bc = idx >> 9, l = idx & 511, b = bc >> 5, c = bc & 31;
  float v = x[((size_t)b * SEQ_LEN + l) * ENC_IN + c];
  xcn[idx] = (v - mean[bc]) / stdv[bc] * rw[c] + rb[c];
}

// ------------------------- Patchify + positional encoding -----------------
__global__ void make_patches(const float* __restrict__ xcn, float* __restrict__ patches,
                             int n_patch, int period, int stride, int total) {
  int idx = blockIdx.x * blockDim.x + threadIdx.x;
  if (idx >= total) return;
  int j = idx % period, rest = idx / period;
  int p = rest % n_patch, bc = rest / n_patch;
  int t = p * stride + j;
  if (t > SEQ_LEN - 1) t = SEQ_LEN - 1;   // right-pad repeats last column
  patches[idx] = xcn[bc * SEQ_LEN + t];
}

__global__ void add_pe(float* __restrict__ z, int n_patch, int total) {
  int idx = blockIdx.x * blockDim.x + threadIdx.x;
  if (idx >= total) return;
  int d = idx & 127, m = idx >> 7, p = m % n_patch;
  int k2 = d & ~1;
  float freq = __expf(-(float)k2 * (9.210340371976184f / 128.0f)); // ln(10000)/d
  float ang  = (float)p * freq;
  z[idx] += (d & 1) ? __cosf(ang) : __sinf(ang);
}

// ------------------------- RMSNorm ----------------------------------------
__global__ void rmsnorm(const float* __restrict__ z, const float* __restrict__ w,
                        float* __restrict__ xn) {
  __shared__ float sred[128];
  int m = blockIdx.x, d = threadIdx.x;
  float v = z[(size_t)m * 128 + d];
  sred[d] = v * v; __syncthreads();
  for (int off = 64; off > 0; off >>= 1) {
    if (d < off) sred[d] += sred[d + off];
    __syncthreads();
  }
  float scale = rsqrtf(sred[0] * (1.0f / 128.0f) + 1e-5f);
  xn[(size_t)m * 128 + d] = v * scale * w[d];
}

// ------------------------- Causal depthwise conv + SiLU -------------------
__global__ void conv_silu(const float* __restrict__ xr, const float* __restrict__ cw,
                          const float* __restrict__ cb, float* __restrict__ ucs,
                          int l, int total) {
  int idx = blockIdx.x * blockDim.x + threadIdx.x;
  if (idx >= total) return;
  int ch = idx & 255, m = idx >> 8;
  int p = m % l, bc = m / l;
  float acc = cb[ch];
#pragma unroll
  for (int t = 0; t < 4; ++t) {
    int pp = p - 3 + t;
    if (pp >= 0) acc += cw[ch * 4 + t] * xr[((size_t)(bc * l + pp)) * 512 + ch];
  }
  ucs[idx] = acc / (1.0f + __expf(-acc));   // SiLU
}

// ------------------------- small GEMM (thread per output) -----------------
// act: 0 = none, 1 = softplus
__global__ void small_gemm(const float* __restrict__ A, const float* __restrict__ W,
                           const float* __restrict__ bias, float* __restrict__ C,
                           int M, int N, int K, int lda, int act) {
  int idx = blockIdx.x * blockDim.x + threadIdx.x;
  if (idx >= M * N) return;
  int n = idx % N, m = idx / N;
  const float* a = A + (size_t)m * lda;
  const float* w = W + (size_t)n * K;
  float acc = bias ? bias[n] : 0.0f;
  for (int k = 0; k < K; ++k) acc += a[k] * w[k];
  if (act == 1) acc = (acc > 20.0f) ? acc : log1pf(__expf(acc));
  C[idx] = acc;
}

// ------------------------- quantum series + softmax -----------------------
__global__ void __launch_bounds__(256)
quantum_softmax(const float* __restrict__ ucs, float* __restrict__ csm,
                int l, float half_inv_mass, float steps) {
  __shared__ float sp[256];
  __shared__ float red[256];
  int m = blockIdx.x, ch = threadIdx.x;
  int p = m % l, bc = m / l, base = bc * l;
  float psi = ucs[(size_t)m * 256 + ch];
  sp[ch] = psi; __syncthreads();
  float prev = sp[(ch + 255) & 255], next = sp[(ch + 1) & 255];
  float kin = -half_inv_mass * (prev - 2.0f * psi + next);
  float pot = 0.0f;
  for (int t = -12; t <= 12; ++t) {
    int pp = p + t; pp = pp < 0 ? 0 : (pp >= l ? l - 1 : pp);
    pot += ucs[(size_t)(base + pp) * 256 + ch];
  }
  pot *= (1.0f / 25.0f);
  float H = kin + pot;
  float prob = psi * psi * __powf(1.0f + H * H, steps);
  red[ch] = prob; __syncthreads();
  for (int off = 128; off > 0; off >>= 1) {
    if (ch < off) red[ch] = fmaxf(red[ch], red[ch + off]);
    __syncthreads();
  }
  float mx = red[0]; __syncthreads();
  float e = __expf(prob - mx);
  red[ch] = e; __syncthreads();
  for (int off = 128; off > 0; off >>= 1) {
    if (ch < off) red[ch] += red[ch + off];
    __syncthreads();
  }
  csm[(size_t)m * 256 + ch] = e / red[0];
}

// ------------------------- selective scan (SSM recurrence) ----------------
// x[p] = sum_{q<=p} dB_u[q] * exp(D[p]-D[q])  ==  h[p] = dB_u[p] + exp(dA[p])*h[p-1]
__global__ void __launch_bounds__(256)
selective_scan(const float* __restrict__ ucs, const float* __restrict__ dt,
               const float* __restrict__ xdbl, const float* __restrict__ Cp,
               const float* __restrict__ A_log, float* __restrict__ y, int l) {
  __shared__ float sB[16], sC[16];
  int bc = blockIdx.x, d = threadIdx.x;
  float Aa[16], h[16];
#pragma unroll
  for (int n = 0; n < 16; ++n) { Aa[n] = -__expf(A_log[d * 16 + n]); h[n] = 0.0f; }
  for (int p = 0; p < l; ++p) {
    int m = bc * l + p;
    if (d < 16) sB[d] = xdbl[(size_t)m * 24 + 8 + d];
    else if (d < 32) sC[d - 16] = Cp[(size_t)m * 16 + (d - 16)];
    __syncthreads();
    float dtv = dt[(size_t)m * 256 + d];
    float du  = dtv * ucs[(size_t)m * 256 + d];
    float acc = 0.0f;
#pragma unroll
    for (int n = 0; n < 16; ++n) {
      h[n] = du * sB[n] + __expf(dtv * Aa[n]) * h[n];
      acc += h[n] * sC[n];
    }
    y[(size_t)m * 256 + d] = acc;
    __syncthreads();
  }
}

// ------------------------- gate, transposes, merge ------------------------
__global__ void gate_silu(const float* __restrict__ xr, float* __restrict__ y, int total) {
  int idx = blockIdx.x * blockDim.x + threadIdx.x;
  if (idx >= total) return;
  int d = idx & 255, m = idx >> 8;
  float res = xr[(size_t)m * 512 + 256 + d];
  y[idx] *= res / (1.0f + __expf(-res));
}

__global__ void head_transpose(const float* __restrict__ z, float* __restrict__ V,
                               int n_patch, int total) {
  int idx = blockIdx.x * blockDim.x + threadIdx.x;  // idx = (bc*128 + d)*n_patch + p
  if (idx >= total) return;
  int p = idx % n_patch, rest = idx / n_patch;
  int d = rest & 127, bc = rest >> 7;
  V[idx] = z[((size_t)(bc * n_patch + p)) * 128 + d];
}

__global__ void zh_transpose(const float* __restrict__ zh, float* __restrict__ zt) {
  int idx = blockIdx.x * blockDim.x + threadIdx.x;  // (b*96+t)*32 + c
  if (idx >= 4 * PRED_LEN * ENC_IN) return;
  int c = idx & 31, rest = idx >> 5, t = rest % PRED_LEN, b = rest / PRED_LEN;
  zt[idx] = zh[((size_t)(b * ENC_IN + c)) * PRED_LEN + t];
}

__global__ void zero_buf(float* __restrict__ p, int n) {
  int i = blockIdx.x * blockDim.x + threadIdx.x;
  if (i < n) p[i] = 0.0f;
}

__global__ void combine_acc(const float* __restrict__ zt, const float* __restrict__ t2,
                            float* __restrict__ fin, float period) {
  int idx = blockIdx.x * blockDim.x + threadIdx.x;
  if (idx >= 4 * PRED_LEN * ENC_IN) return;
  fin[idx] += period * (zt[idx] + t2[idx]);
}

__global__ void finalize(const float* __restrict__ fin, const float* __restrict__ rw,
                         const float* __restrict__ rb, const float* __restrict__ mean,
                         const float* __restrict__ stdv, float* __restrict__ out) {
  int idx = blockIdx.x * blockDim.x + threadIdx.x;
  if (idx >= 4 * PRED_LEN * ENC_IN) return;
  int c = idx & 31, rest = idx >> 5, b = rest / PRED_LEN;
  int bc = b * ENC_IN + c;
  float v = fin[idx] * (1.0f / 48.0f);            // / sum(PERIODS)
  v = (v - rb[c]) / (rw[c] + 1e-10f);             // EPS*EPS
  out[idx] = v * stdv[bc] + mean[bc];
}

// ===========================================================================
extern "C" void kernel_launch(void* const* d_in, const int* in_sizes, int n_in,
                              void* d_out, int out_size, void* d_ws, size_t ws_size,
                              hipStream_t stream) {
  (void)in_sizes; (void)n_in; (void)out_size; (void)ws_size;
  const float* x_enc = (const float*)d_in[0];
  // params flattened in jax-tree order (sorted dict keys), after 4 inputs:
  // blocks[2][2] x {A_log,C_proj_w,conv_b,conv_w,dt_proj_b,dt_proj_w,in_proj_w,norm_w,out_proj_w,x_proj_w}
  const float* P[2][2][10];
  for (int i = 0; i < 2; ++i)
    for (int j = 0; j < 2; ++j)
      for (int k = 0; k < 10; ++k)
        P[i][j][k] = (const float*)d_in[4 + (i * 2 + j) * 10 + k];
  const float* ch_b1 = (const float*)d_in[44];
  const float* ch_b2 = (const float*)d_in[45];
  const float* ch_w1 = (const float*)d_in[46];
  const float* ch_w2 = (const float*)d_in[47];
  const float* head_b[2] = {(const float*)d_in[48], (const float*)d_in[50]};
  const float* head_w[2] = {(const float*)d_in[49], (const float*)d_in[51]};
  const float* patch_w[2] = {(const float*)d_in[52], (const float*)d_in[53]};
  const float* revin_b = (const float*)d_in[54];
  const float* revin_w = (const float*)d_in[55];
  float* out = (float*)d_out;

  // workspace layout (floats), sized for the smaller period (M = 8192)
  float* ws      = (float*)d_ws;
  float* mean    = ws;
  float* stdv    = ws + 128;
  float* xcn     = ws + 256;            // 128*512
  float* patches = xcn + 65536;         // 8192*16 / 4096*32
  float* z       = patches + 131072;    // 8192*128
  float* xn      = z + 1048576;         // 8192*128
  float* xr      = xn + 1048576;        // 8192*512
  float* ucs     = xr + 4194304;        // 8192*256
  float* xdbl    = ucs + 2097152;       // 8192*24
  float* csm     = xdbl + 196608;       // 8192*256
  float* cp      = csm + 2097152;       // 8192*16
  float* dtb     = cp + 131072;         // 8192*256
  float* ybuf    = dtb + 2097152;       // 8192*256
  float* V       = ybuf + 2097152;      // 128*8192
  float* zh      = V + 1048576;         // 128*96
  float* zt      = zh + 12288;          // 384*32
  float* hbuf    = zt + 12288;          // 384*128
  float* t2      = hbuf + 49152;        // 384*32
  float* fin     = t2 + 12288;          // 384*32

  auto cdiv = [](int a, int b) { return (a + b - 1) / b; };
  const int NOUT = 4 * PRED_LEN * ENC_IN;  // 12288

  zero_buf<<<cdiv(NOUT, 256), 256, 0, stream>>>(fin, NOUT);
  revin_stats<<<128, 128, 0, stream>>>(x_enc, mean, stdv);
  revin_apply<<<cdiv(128 * SEQ_LEN, 256), 256, 0, stream>>>(x_enc, mean, stdv,
                                                            revin_w, revin_b, xcn);

  const int periods[2] = {16, 32};
  for (int i = 0; i < 2; ++i) {
    const int period = periods[i], stride = period / 2;
    const int n_patch = SEQ_LEN / stride;       // 64 / 32
    const int M = 128 * n_patch;                // 8192 / 4096

    make_patches<<<cdiv(M * period, 256), 256, 0, stream>>>(xcn, patches, n_patch,
                                                            period, stride, M * period);
    gemm_wmma<4, 0, 0, 0><<<dim3(M / 16, D_MODEL / 64), 32, 0, stream>>>(
        patches, patch_w[i], nullptr, z, M, D_MODEL, period);
    add_pe<<<cdiv(M * D_MODEL, 256), 256, 0, stream>>>(z, n_patch, M * D_MODEL);

    for (int j = 0; j < 2; ++j) {
      const float* A_log  = P[i][j][0];
      const float* C_proj = P[i][j][1];
      const float* conv_b = P[i][j][2];
      const float* conv_w = P[i][j][3];
      const float* dt_b   = P[i][j][4];
      const float* dt_w   = P[i][j][5];
      const float* in_w   = P[i][j][6];
      const float* norm_w = P[i][j][7];
      const float* out_w  = P[i][j][8];
      const float* x_w    = P[i][j][9];

      rmsnorm<<<M, 128, 0, stream>>>(z, norm_w, xn);
      gemm_wmma<4, 0, 0, 0><<<dim3(M / 16, 512 / 64), 32, 0, stream>>>(
          xn, in_w, nullptr, xr, M, 512, 128);
      conv_silu<<<cdiv(M * 256, 256), 256, 0, stream>>>(xr, conv_w, conv_b, ucs,
                                                        n_patch, M * 256);
      small_gemm<<<cdiv(M * 24, 128), 128, 0, stream>>>(ucs, x_w, nullptr, xdbl,
                                                        M, 24, 256, 256, 0);
      quantum_softmax<<<M, 256, 0, stream>>>(ucs, csm, n_patch,
                                             0.5f / (float)period, (float)n_patch);
      small_gemm<<<cdiv(M * 16, 128), 128, 0, stream>>>(csm, C_proj, nullptr, cp,
                                                        M, 16, 256, 256, 0);
      small_gemm<<<cdiv(M * 256, 128), 128, 0, stream>>>(xdbl, dt_w, dt_b, dtb,
                                                         M, 256, 8, 24, 1);
      selective_scan<<<128, 256, 0, stream>>>(ucs, dtb, xdbl, cp, A_log, ybuf, n_patch);
      gate_silu<<<cdiv(M * 256, 256), 256, 0, stream>>>(xr, ybuf, M * 256);
      gemm_wmma<4, 0, 1, 0><<<dim3(M / 16, D_MODEL / 64), 32, 0, stream>>>(
          ybuf, out_w, nullptr, z, M, D_MODEL, 256);   // residual accumulate
    }

    const int nf = D_MODEL * n_patch;   // 8192 / 4096
    head_transpose<<<cdiv(128 * nf, 256), 256, 0, stream>>>(z, V, n_patch, 128 * nf);
    gemm_wmma<2, 1, 0, 0><<<dim3(128 / 16, PRED_LEN / 32), 32, 0, stream>>>(
        V, head_w[i], head_b[i], zh, 128, PRED_LEN, nf);
    zh_transpose<<<cdiv(NOUT, 256), 256, 0, stream>>>(zh, zt);
    gemm_wmma<4, 1, 0, 1><<<dim3(384 / 16, 128 / 64), 32, 0, stream>>>(
        zt, ch_w1, ch_b1, hbuf, 384, 128, 32);          // + GELU fused
    gemm_wmma<2, 1, 0, 0><<<dim3(384 / 16, 32 / 32), 32, 0, stream>>>(
        hbuf, ch_w2, ch_b2, t2, 384, 32, 128);
    combine_acc<<<cdiv(NOUT, 256), 256, 0, stream>>>(zt, t2, fin, (float)period);
  }

  finalize<<<cdiv(NOUT, 256), 256, 0, stream>>>(fin, revin_w, revin_b, mean, stdv, out);
}